// DTW_20675972563812
// MI455X (gfx1250) — compile-verified
//
#include <hip/hip_runtime.h>
#include <math.h>

// CDNA5 / gfx1250: wave32, WMMA 16x16x4 f32 (K=4 matches the C=4 channel dim).
typedef __attribute__((ext_vector_type(2))) float v2f;  // A/B frag: 16x4 / 4x16 f32 = 2 VGPRs
typedef __attribute__((ext_vector_type(8))) float v8f;  // C/D frag: 16x16 f32 = 8 VGPRs

#define TLEN   1024
#define NTRACE 32
#define WIN    100
#define NTHR   256   // 8 wave32 waves
#define DSTR   257   // dtile row stride (pad to dodge LDS bank conflicts)

// Raw hardware sqrt (v_sqrt_f32, ~1 ulp) — skips the IEEE range-scaling +
// Newton fixup sequence the libm sqrtf lowering emits. Inputs here are sums
// of squares of N(0,1) data: always well inside the normal range.
__device__ __forceinline__ float fast_sqrtf(float v) {
    return __builtin_amdgcn_sqrtf(v);
}

__launch_bounds__(NTHR, 1)
__global__ void dtw_banded_wmma(const float* __restrict__ xg,
                                const float* __restrict__ yg,
                                float* __restrict__ out) {
    __shared__ float xs[TLEN * 4];     // trace x, (t,c) row-major
    __shared__ float ys[TLEN * 4];     // trace y
    __shared__ float xn[TLEN];         // ||x_t||^2
    __shared__ float yn[TLEN];         // ||y_s||^2
    __shared__ float rowbuf[TLEN];     // current DP row (full width, zeros outside band)
    __shared__ float dtile[16 * DSTR]; // 16-row distance slab covering the band
    __shared__ float wTotA[8], wTotB[8];

    const int trace = blockIdx.x;
    const int tid   = threadIdx.x;
    const int lane  = tid & 31;
    const int wid   = tid >> 5;

    // ---- Stage trace into LDS: x/y layout is (t, n, c) so each t is one float4 ----
    for (int t = tid; t < TLEN; t += NTHR) {
        const float4 vx = *reinterpret_cast<const float4*>(xg + (size_t)(t * NTRACE + trace) * 4);
        xs[t * 4 + 0] = vx.x; xs[t * 4 + 1] = vx.y; xs[t * 4 + 2] = vx.z; xs[t * 4 + 3] = vx.w;
        xn[t] = vx.x * vx.x + vx.y * vx.y + vx.z * vx.z + vx.w * vx.w;
        const float4 vy = *reinterpret_cast<const float4*>(yg + (size_t)(t * NTRACE + trace) * 4);
        ys[t * 4 + 0] = vy.x; ys[t * 4 + 1] = vy.y; ys[t * 4 + 2] = vy.z; ys[t * 4 + 3] = vy.w;
        yn[t] = vy.x * vy.x + vy.y * vy.y + vy.z * vy.z + vy.w * vy.w;
        rowbuf[t] = 0.0f;
    }
    __syncthreads();

    for (int rb = 0; rb < 64; ++rb) {          // 16-row blocks
        const int i0   = rb * 16;
        const int ctLo = (rb - 7 > 0) ? (rb - 7) : 0;
        const int ctHi = (rb + 7 < 63) ? (rb + 7) : 63;
        const int c0   = ctLo * 16;            // first column held in dtile

        // ---- Phase 1: distance slab via V_WMMA_F32_16X16X4_F32 ----
        {
            const int m    = lane & 15;        // M (A) / N (B) index
            const int koff = (lane >> 4) * 2;  // lanes 16..31 hold K=2,3 (ISA 16x4 f32 layout)
            v2f afrag;
            afrag.x = xs[(i0 + m) * 4 + koff];
            afrag.y = xs[(i0 + m) * 4 + koff + 1];
            for (int ct = ctLo + wid; ct <= ctHi; ct += 8) {   // 8 waves split <=15 tiles
                const int j0 = ct * 16;
                v2f bfrag;
                bfrag.x = ys[(j0 + m) * 4 + koff];
                bfrag.y = ys[(j0 + m) * 4 + koff + 1];
                v8f acc = {};
                acc = __builtin_amdgcn_wmma_f32_16x16x4_f32(
                    /*neg_a=*/false, afrag, /*neg_b=*/false, bfrag,
                    /*c_mod=*/(short)0, acc, /*reuse_a=*/false, /*reuse_b=*/false);
                // D layout: VGPR r -> M=r (lanes 0-15) / M=r+8 (lanes 16-31), N = lane%16
                const int   half8 = (lane >> 4) * 8;
                const float ynj   = yn[j0 + m];
                #pragma unroll
                for (int r = 0; r < 8; ++r) {
                    const int   row = r + half8;
                    const float sq  = xn[i0 + row] + ynj - 2.0f * acc[r];
                    dtile[row * DSTR + (j0 - c0) + m] = fast_sqrtf(fmaxf(sq, 0.0f));
                }
            }
        }
        __syncthreads();

        int rStart = 0;
        if (rb == 0) {
            // acc[0, j] = d[0, j]; only j<=100 is ever read before being zeroed/overwritten
            for (int j = tid; j <= WIN; j += NTHR) rowbuf[j] = dtile[j];  // row 0, c0==0
            rStart = 1;
            __syncthreads();
        }

        // ---- Phase 2: DP rows, (min,+) parallel scan across the band ----
        for (int r = rStart; r < 16; ++r) {
            const int  i     = i0 + r;
            const int  js    = (i - WIN > 1) ? (i - WIN) : 1;
            const int  je    = (i + WIN < TLEN) ? (i + WIN) : TLEN;
            const int  width = je - js;        // <= 200, so <= 7 active waves
            const int  j     = js + tid;
            const bool act   = tid < width;

            float a = 0.0f, b = 1e30f;         // identity transform for idle lanes
            if (act) {
                const float dij = dtile[r * DSTR + (j - c0)];
                const float m2  = fminf(rowbuf[j - 1], rowbuf[j]);   // prev-row neighbors
                a = dij;
                b = dij + m2;
            }
            // intra-wave inclusive scan: compose((Au,Bu) earlier, (A,B) later)
            float A = a, B = b;
            #pragma unroll
            for (int d = 1; d < 32; d <<= 1) {
                const float Au = __shfl_up(A, d, 32);
                const float Bu = __shfl_up(B, d, 32);
                if (lane >= d) { B = fminf(B, Bu + A); A += Au; }
            }
            if (lane == 31) { wTotA[wid] = A; wTotB[wid] = B; }
            __syncthreads();                    // also fences all rowbuf reads above

            // cross-wave exclusive prefix (<=7 iterations, redundant per lane)
            float PA = 0.0f, PB = 1e30f;
            for (int w = 0; w < wid; ++w) {
                PB = fminf(wTotB[w], PB + wTotA[w]);
                PA += wTotA[w];
            }
            const float FA = PA + A;
            const float FB = fminf(B, PB + A);
            // chain seed: acc[i,0]=d[i,0] when the band starts at j=1 (then i<=101 => c0==0),
            // otherwise the out-of-band zero to the left of the band.
            const float xinit = (js == 1) ? dtile[r * DSTR] : 0.0f;
            const float xv    = fminf(FB, xinit + FA);

            if (act) rowbuf[j] = xv;
            if (tid == 254) {                   // cell leaving the band becomes 0 (torch quirk)
                const int jz = i - WIN - 1;
                if (jz >= 1) rowbuf[jz] = 0.0f;
            }
            if (tid == 255) {                   // acc[i,0] = d[i,0]
                const float d0 = xs[i * 4 + 0] - ys[0];
                const float d1 = xs[i * 4 + 1] - ys[1];
                const float d2 = xs[i * 4 + 2] - ys[2];
                const float d3 = xs[i * 4 + 3] - ys[3];
                rowbuf[0] = fast_sqrtf(d0 * d0 + d1 * d1 + d2 * d2 + d3 * d3);
            }
            __syncthreads();
        }
    }

    if (tid == 0) atomicAdd(out, rowbuf[TLEN - 1] * (1.0f / (float)NTRACE));
}

extern "C" void kernel_launch(void* const* d_in, const int* in_sizes, int n_in,
                              void* d_out, int out_size, void* d_ws, size_t ws_size,
                              hipStream_t stream) {
    (void)in_sizes; (void)n_in; (void)out_size; (void)d_ws; (void)ws_size;
    const float* x = (const float*)d_in[0];
    const float* y = (const float*)d_in[1];
    float* out = (float*)d_out;
    hipMemsetAsync(out, 0, sizeof(float), stream);   // graph-capture safe
    dtw_banded_wmma<<<dim3(NTRACE), dim3(NTHR), 0, stream>>>(x, y, out);
}